// UltraSparseGNNBaseline_49409303773458
// MI455X (gfx1250) — compile-verified
//
#include <hip/hip_runtime.h>

typedef __attribute__((ext_vector_type(2))) float v2f;
typedef __attribute__((ext_vector_type(8))) float v8f;

#define NS   16      // species / nodes per graph
#define DH   32      // hidden dim
#define WPB  8       // waves per block (1 wave = 1 graph)
#define HSTR 34      // padded LDS row stride (floats) for 16x32 buffers (even -> b64-aligned, bank-conflict-free)
#define ASTR 18      // padded LDS row stride for attn 16x16
#define H_OFF  0
#define HW_OFF (16 * HSTR)
#define AT_OFF (2 * 16 * HSTR)
#define S_OFF  (2 * 16 * HSTR + 16 * ASTR)
#define WAVE_F (S_OFF + 32)

__device__ __forceinline__ v8f wmma_f32_k4(v2f a, v2f b, v8f c) {
  // D = A(16x4) * B(4x16) + C(16x16), all f32
  return __builtin_amdgcn_wmma_f32_16x16x4_f32(
      /*neg_a=*/false, a, /*neg_b=*/false, b,
      /*c_mod=*/(short)0, c, /*reuse_a=*/false, /*reuse_b=*/false);
}

__global__ __launch_bounds__(256, 1) void ultra_gat_kernel(
    const float* __restrict__ x,
    const float* __restrict__ w_in,
    const float* __restrict__ emb,
    const float* __restrict__ gw0,
    const float* __restrict__ ga0,
    const float* __restrict__ gw1,
    const float* __restrict__ ga1,
    const float* __restrict__ w_out,
    const float* __restrict__ b_out,
    const float* __restrict__ r_int,
    float* __restrict__ out,
    int n_graphs)
{
  __shared__ float smem[WPB * WAVE_F];
  const int lane = threadIdx.x & 31;
  const int wave = threadIdx.x >> 5;
  const int g    = blockIdx.x * WPB + wave;
  if (g >= n_graphs) return;              // uniform per wave

  float* hs  = smem + wave * WAVE_F + H_OFF;   // h  : 16 x 32, row-major, stride HSTR
  float* hws = smem + wave * WAVE_F + HW_OFF;  // hw : 16 x 32, row-major, stride HSTR
  float* ats = smem + wave * WAVE_F + AT_OFF;  // attn: 16 x 16, stride ASTR
  float* ss  = smem + wave * WAVE_F + S_OFF;   // s_i (0..15), s_j (16..31)

  const int hi   = lane >> 4;    // lane half: 0 => lanes 0-15, 1 => lanes 16-31
  const int l15  = lane & 15;
  const int koff = 2 * hi;       // K offset within a K=4 chunk for f32 WMMA operands

  // -------- input projection: h[n][d] = x*w_in[d][0] + log(x)*w_in[d][1] + emb[n][d]
  {
    const int n  = lane >> 1;
    const int d0 = (lane & 1) * 16;
    const float xv = x[g * NS + n];
    const float lv = __logf(fmaxf(xv, 1e-6f));
    #pragma unroll
    for (int i = 0; i < 16; ++i) {
      const int d = d0 + i;
      hs[n * HSTR + d] = fmaf(xv, w_in[2 * d], fmaf(lv, w_in[2 * d + 1], emb[n * DH + d]));
    }
  }

  #pragma unroll
  for (int layer = 0; layer < 2; ++layer) {
    const float* gw = layer ? gw1 : gw0;
    const float* ga = layer ? ga1 : ga0;

    // -------- hw = h @ W^T : two 16x16 N-tiles, K=32 as 8 chunks of 4
    // A operand (ISA layout): vgpr pair c holds h[row][4c+koff], h[row][4c+koff+1]
    v2f a[8];
    #pragma unroll
    for (int c = 0; c < 8; ++c)
      a[c] = *(const v2f*)(hs + l15 * HSTR + 4 * c + koff);

    v8f acc0 = {}; v8f acc1 = {};
    #pragma unroll
    for (int c = 0; c < 8; ++c) {
      // B[k][n'] = W[n'][k]; lane col n' = l15 (+16 for tile1); rows 4c+koff, +1 are contiguous
      v2f b0 = *(const v2f*)(gw + (l15)      * DH + 4 * c + koff);
      v2f b1 = *(const v2f*)(gw + (l15 + 16) * DH + 4 * c + koff);
      acc0 = wmma_f32_k4(a[c], b0, acc0);
      acc1 = wmma_f32_k4(a[c], b1, acc1);
    }

    // scatter hw (C/D layout) -> LDS row-major
    #pragma unroll
    for (int r = 0; r < 8; ++r) {
      hws[(r + 8 * hi) * HSTR + l15]      = acc0[r];
      hws[(r + 8 * hi) * HSTR + 16 + l15] = acc1[r];
    }

    // -------- scores: lanes 0-15 compute s_i[n]=hw[n].a[:32], lanes 16-31 s_j[m]=hw[m].a[32:]
    {
      const float* av = ga + DH * hi;
      const float* hr = hws + l15 * HSTR;
      float s = 0.f;
      #pragma unroll
      for (int d = 0; d < DH; ++d) s = fmaf(hr[d], av[d], s);
      ss[lane] = s;
    }

    // -------- per-row top-4 mask + softmax (one row per lane 0-15)
    if (hi == 0) {
      const float si = ss[l15];
      float sc[16];
      #pragma unroll
      for (int m = 0; m < 16; ++m) sc[m] = si + ss[16 + m];
      float t0 = -3.4e38f, t1 = t0, t2 = t0, t3 = t0;   // running top-4
      #pragma unroll
      for (int m = 0; m < 16; ++m) {
        const float v = sc[m];
        if (v > t0)      { t3 = t2; t2 = t1; t1 = t0; t0 = v; }
        else if (v > t1) { t3 = t2; t2 = t1; t1 = v; }
        else if (v > t2) { t3 = t2; t2 = v; }
        else if (v > t3) { t3 = v; }
      }
      float e[16];
      float sum = 0.f;
      #pragma unroll
      for (int m = 0; m < 16; ++m) {
        const float v  = sc[m];
        const float ev = (v >= t3) ? __expf(v - t0) : 0.f;  // non-top-4 -> -inf -> 0
        e[m] = ev; sum += ev;
      }
      const float inv = 1.0f / sum;
      #pragma unroll
      for (int m = 0; m < 16; ++m) ats[l15 * ASTR + m] = e[m] * inv;
    }

    // -------- h = h + attn @ hw  (residual folded into C operand), K=16 as 4 chunks
    v8f c0, c1;
    #pragma unroll
    for (int r = 0; r < 8; ++r) {         // gather current h in C layout
      c0[r] = hs[(r + 8 * hi) * HSTR + l15];
      c1[r] = hs[(r + 8 * hi) * HSTR + 16 + l15];
    }
    #pragma unroll
    for (int c = 0; c < 4; ++c) {
      v2f aa = *(const v2f*)(ats + l15 * ASTR + 4 * c + koff);
      v2f b0, b1;
      b0.x = hws[(4 * c + koff)     * HSTR + l15];        // B[k=m][n=d] = hw[m][d]
      b0.y = hws[(4 * c + koff + 1) * HSTR + l15];
      b1.x = hws[(4 * c + koff)     * HSTR + 16 + l15];
      b1.y = hws[(4 * c + koff + 1) * HSTR + 16 + l15];
      c0 = wmma_f32_k4(aa, b0, c0);
      c1 = wmma_f32_k4(aa, b1, c1);
    }
    #pragma unroll
    for (int r = 0; r < 8; ++r) {         // write back new h, row-major
      hs[(r + 8 * hi) * HSTR + l15]      = c0[r];
      hs[(r + 8 * hi) * HSTR + 16 + l15] = c1[r];
    }
  }

  // -------- output projection: out[g][n] = h[n].w_out + b_out + r_intrinsic[n]
  if (hi == 0) {
    const float* hr = hs + l15 * HSTR;
    float acc = b_out[0] + r_int[l15];
    #pragma unroll
    for (int d = 0; d < DH; ++d) acc = fmaf(hr[d], w_out[d], acc);
    out[g * NS + l15] = acc;
  }
}

extern "C" void kernel_launch(void* const* d_in, const int* in_sizes, int n_in,
                              void* d_out, int out_size, void* d_ws, size_t ws_size,
                              hipStream_t stream) {
  const float* x    = (const float*)d_in[0];
  const float* w_in = (const float*)d_in[1];
  const float* emb  = (const float*)d_in[2];
  const float* gw0  = (const float*)d_in[3];
  const float* ga0  = (const float*)d_in[4];
  const float* gw1  = (const float*)d_in[5];
  const float* ga1  = (const float*)d_in[6];
  const float* wout = (const float*)d_in[7];
  const float* bout = (const float*)d_in[8];
  const float* rint = (const float*)d_in[9];
  float* out = (float*)d_out;

  const int n_graphs = in_sizes[0] / NS;          // B*T = 65536
  const int blocks   = (n_graphs + WPB - 1) / WPB;

  hipLaunchKernelGGL(ultra_gat_kernel, dim3(blocks), dim3(256), 0, stream,
                     x, w_in, emb, gw0, ga0, gw1, ga1, wout, bout, rint,
                     out, n_graphs);
}